// MemSpecialist_60576218742845
// MI455X (gfx1250) — compile-verified
//
#include <hip/hip_runtime.h>
#include <stdint.h>

// Sizes from the reference
#define BATCH 256
#define DDIM  512
#define HDIM  2048
#define NSLOT 32768
#define NBLK  128     // persistent grid: 128 blocks x 256 thr (8 waves) -> co-resident
#define TPB   256

typedef float v2f __attribute__((ext_vector_type(2)));
typedef float v8f __attribute__((ext_vector_type(8)));

// ---------- packed (score, slot) for first-max argmax via atomic_max_u64 ----------
__device__ __forceinline__ unsigned orderedF(float f) {
  unsigned u = __float_as_uint(f);
  return (u & 0x80000000u) ? ~u : (u | 0x80000000u);
}
__device__ __forceinline__ float unorderedF(unsigned of) {
  unsigned u = (of & 0x80000000u) ? (of ^ 0x80000000u) : ~of;
  return __uint_as_float(u);
}

// ---------- WMMA helper: 16 dot-products (rows of A  x  broadcast vector) ----------
// A: row-major [*, lda]; computes D[m][n] = sum_k A[row_base+m][k] * vec[k] for all n
// (vec broadcast across the 16 B-columns). Result: acc[r] = dot(row_base + half*8 + r)
// replicated across lanes of each half-wave.
// f32 WMMA layout: A 16x4: lanes0-15 M=0..15 K=0,1; lanes16-31 M=0..15 K=2,3.
//                  B 4x16:  lanes0-15 rows K=0,1;   lanes16-31 rows K=2,3.
__device__ __forceinline__ v8f wmma_rows_vec(const float* __restrict__ A, size_t lda,
                                             int row_base, const float* __restrict__ vec,
                                             int K) {
  const int lane = threadIdx.x & 31;
  const int half = lane >> 4;
  const float* ap = A + (size_t)(row_base + (lane & 15)) * lda + 2 * half;
  const float* bp = vec + 2 * half;
  v8f acc = {0.f, 0.f, 0.f, 0.f, 0.f, 0.f, 0.f, 0.f};
  __builtin_prefetch(ap, 0, 1);   // -> global_prefetch_b8
#pragma unroll 4
  for (int c = 0; c < K; c += 4) {
    v2f a = *(const v2f*)(ap + c);
    v2f b = *(const v2f*)(bp + c);
    acc = __builtin_amdgcn_wmma_f32_16x16x4_f32(false, a, false, b, (short)0, acc,
                                                false, false);
  }
  return acc;
}

// ---------- device-scope grid barrier (generation counter) ----------
__device__ __forceinline__ void gbar(unsigned* cnt, unsigned* gen) {
  __threadfence();
  __syncthreads();
  if (threadIdx.x == 0) {
    unsigned g = __hip_atomic_load(gen, __ATOMIC_RELAXED, __HIP_MEMORY_SCOPE_AGENT);
    unsigned a = __hip_atomic_fetch_add(cnt, 1u, __ATOMIC_ACQ_REL, __HIP_MEMORY_SCOPE_AGENT);
    if (a == (unsigned)(NBLK - 1)) {
      __hip_atomic_store(cnt, 0u, __ATOMIC_RELAXED, __HIP_MEMORY_SCOPE_AGENT);
      __hip_atomic_fetch_add(gen, 1u, __ATOMIC_RELEASE, __HIP_MEMORY_SCOPE_AGENT);
    } else {
      while (__hip_atomic_load(gen, __ATOMIC_ACQUIRE, __HIP_MEMORY_SCOPE_AGENT) == g)
        __builtin_amdgcn_s_sleep(2);
    }
  }
  __syncthreads();
}

// =====================================================================
// Kernel 1: precompute ALL key/value projections as a real WMMA GEMM:
//   Kall  = S_t @ Wk.T + bk   [256 x 512]
//   WVall = S_t @ Wv.T + bv   [256 x 512]
// 1024 16x16 output tiles -> 128 blocks x 8 waves, one tile per wave.
// =====================================================================
__global__ __launch_bounds__(TPB)
void proj_gemm(const float* __restrict__ S, const float* __restrict__ Wk,
               const float* __restrict__ bk, const float* __restrict__ Wv,
               const float* __restrict__ bv, float* __restrict__ Kall,
               float* __restrict__ WVall) {
  const int lane = threadIdx.x & 31;
  const int wid = threadIdx.x >> 5;
  int job = blockIdx.x * 8 + wid;  // 0..1023
  const float* W;
  const float* b;
  float* O;
  if (job < 512) { W = Wk; b = bk; O = Kall; }
  else           { W = Wv; b = bv; O = WVall; job -= 512; }
  const int m = (job >> 5) * 16;   // 16 m-tiles (256 rows)
  const int n = (job & 31) * 16;   // 32 n-tiles (512 cols)
  const int half = lane >> 4;
  // A = rows of S (M,K);  B.T rows = rows of W (because out = S @ W.T)
  const float* ap = S + (size_t)(m + (lane & 15)) * DDIM + 2 * half;
  const float* bp = W + (size_t)(n + (lane & 15)) * DDIM + 2 * half;
  v8f acc = {0.f, 0.f, 0.f, 0.f, 0.f, 0.f, 0.f, 0.f};
#pragma unroll 4
  for (int c = 0; c < DDIM; c += 4) {
    v2f a = *(const v2f*)(ap + c);
    v2f bb = *(const v2f*)(bp + c);
    acc = __builtin_amdgcn_wmma_f32_16x16x4_f32(false, a, false, bb, (short)0, acc,
                                                false, false);
  }
  const int col = n + (lane & 15);
  const float bias = b[col];
#pragma unroll
  for (int r = 0; r < 8; r++)
    O[(size_t)(m + half * 8 + r) * DDIM + col] = acc[r] + bias;
}

// =====================================================================
// Kernel 2: persistent scan over the 256 steps, grid-synced phases:
//  P1: scores = mem_keys @ k  (WMMA chains, 32 slots/wave) + global max/argmax
//  P2: w=exp(scores-max); wsum; rv += w @ mem_vals (coalesced VALU stream)
//  P3: scatter write mk/mv[slot]; h = relu(W1 @ [s_i, rv/wsum] + b1)  (WMMA)
//  P4: delta = W2 @ h + b2 -> out[i]                                  (WMMA)
// =====================================================================
__global__ __launch_bounds__(TPB, 1)
void memsim_persistent(float* __restrict__ mk, float* __restrict__ mv,
                       const float* __restrict__ Kall, const float* __restrict__ WVall,
                       const float* __restrict__ S, const float* __restrict__ W1,
                       const float* __restrict__ b1, const float* __restrict__ W2,
                       const float* __restrict__ b2, float* __restrict__ scores,
                       float* __restrict__ rv, float* __restrict__ h,
                       float* __restrict__ wsum, unsigned long long* __restrict__ maxpack,
                       unsigned* cnt, unsigned* gen, float* __restrict__ out) {
  __shared__ __align__(16) float lds[HDIM];  // 8 KB, reused per phase
  const int tid = threadIdx.x;
  const int lane = tid & 31;
  const int wid = tid >> 5;
  const int blk = blockIdx.x;

  for (int i = 0; i < BATCH; i++) {
    // ---------------- P1: scores + max/argmax ----------------
    for (int j = tid; j < DDIM; j += TPB) lds[j] = Kall[(size_t)i * DDIM + j];
    __syncthreads();
    unsigned long long mp;
    {
#pragma unroll
      for (int ch = 0; ch < 2; ch++) {
        const int sbase = blk * 256 + (wid * 2 + ch) * 16;  // 256 slots per block
        v8f acc = wmma_rows_vec(mk, DDIM, sbase, lds, DDIM);
        const int half = lane >> 4;
        if ((lane & 15) == 0) {
#pragma unroll
          for (int r = 0; r < 8; r++) scores[sbase + half * 8 + r] = acc[r];
        }
        unsigned long long best = 0ull;
#pragma unroll
        for (int r = 0; r < 8; r++) {
          const int s = sbase + half * 8 + r;
          unsigned long long p =
              ((unsigned long long)orderedF(acc[r]) << 32) | (unsigned)(~(unsigned)s);
          best = (p > best) ? p : best;
        }
        unsigned long long o = __shfl_xor(best, 16, 32);
        best = (o > best) ? o : best;
        if (lane == 0) atomicMax(maxpack, best);
      }
    }
    gbar(cnt, gen);

    // ---------------- P2: softmax weights + rv accumulation ----------------
    mp = __hip_atomic_load(maxpack, __ATOMIC_RELAXED, __HIP_MEMORY_SCOPE_AGENT);
    const float maxv = unorderedF((unsigned)(mp >> 32));
    const int base = blk * 256;
    lds[tid] = __expf(scores[base + tid] - maxv);
    __syncthreads();
    if (wid == 0) {  // block-partial wsum
      float s = 0.f;
      for (int j = lane; j < 256; j += 32) s += lds[j];
#pragma unroll
      for (int o = 16; o > 0; o >>= 1) s += __shfl_xor(s, o, 32);
      if (lane == 0)
        __hip_atomic_fetch_add(wsum, s, __ATOMIC_RELAXED, __HIP_MEMORY_SCOPE_AGENT);
    }
    {  // each thread owns 2 contiguous columns -> fully coalesced 2KB/row stream
      float ax = 0.f, ay = 0.f;
      const float* mvp = mv + (size_t)base * DDIM + 2 * tid;
      for (int s = 0; s < 256; s++) {
        const float w = lds[s];
        v2f m = *(const v2f*)(mvp + (size_t)s * DDIM);
        ax += w * m[0];
        ay += w * m[1];
      }
      __hip_atomic_fetch_add(&rv[2 * tid], ax, __ATOMIC_RELAXED, __HIP_MEMORY_SCOPE_AGENT);
      __hip_atomic_fetch_add(&rv[2 * tid + 1], ay, __ATOMIC_RELAXED,
                             __HIP_MEMORY_SCOPE_AGENT);
    }
    gbar(cnt, gen);

    // ---------------- P3: memory write + hidden layer ----------------
    const int slot = (int)(~(unsigned)(mp & 0xffffffffull));
    if (blk == 0) {
      for (int j = tid; j < DDIM; j += TPB) {
        mk[(size_t)slot * DDIM + j] = Kall[(size_t)i * DDIM + j];
        mv[(size_t)slot * DDIM + j] = WVall[(size_t)i * DDIM + j];
      }
    }
    const float inv =
        1.f / __hip_atomic_load(wsum, __ATOMIC_RELAXED, __HIP_MEMORY_SCOPE_AGENT);
    for (int j = tid; j < DDIM; j += TPB) {
      lds[j] = S[(size_t)i * DDIM + j];     // merged[0:512]   = s_i
      lds[DDIM + j] = rv[j] * inv;          // merged[512:1024] = read_val
    }
    __syncthreads();
    if (wid == 0) {  // 2048 hidden outputs = 128 blocks x 16-row WMMA chain
      const int obase = blk * 16;
      v8f acc = wmma_rows_vec(W1, 2 * DDIM, obase, lds, 2 * DDIM);
      const int half = lane >> 4;
      if ((lane & 15) == 0) {
#pragma unroll
        for (int r = 0; r < 8; r++) {
          const int o = obase + half * 8 + r;
          const float v = acc[r] + b1[o];
          h[o] = v > 0.f ? v : 0.f;  // relu
        }
      }
    }
    gbar(cnt, gen);

    // ---------------- P4: output layer + per-step state reset ----------------
    if (blk < 32) {  // 512 outputs = 32 blocks x 16-row WMMA chain
      for (int j = tid; j < HDIM; j += TPB) lds[j] = h[j];
      __syncthreads();
      if (wid == 0) {
        const int obase = blk * 16;
        v8f acc = wmma_rows_vec(W2, HDIM, obase, lds, HDIM);
        const int half = lane >> 4;
        if ((lane & 15) == 0) {
#pragma unroll
          for (int r = 0; r < 8; r++) {
            const int o = obase + half * 8 + r;
            out[(size_t)i * DDIM + o] = acc[r] + b2[o];
          }
        }
      }
    }
    if (blk == 64)
      for (int j = tid; j < DDIM; j += TPB) rv[j] = 0.f;
    if (blk == 65 && tid == 0) *wsum = 0.f;
    if (blk == 66 && tid == 0) *maxpack = 0ull;
    gbar(cnt, gen);
  }
}

// =====================================================================
extern "C" void kernel_launch(void* const* d_in, const int* in_sizes, int n_in,
                              void* d_out, int out_size, void* d_ws, size_t ws_size,
                              hipStream_t stream) {
  (void)in_sizes; (void)n_in; (void)out_size; (void)ws_size;
  const float* S   = (const float*)d_in[0];
  const float* mk0 = (const float*)d_in[1];
  const float* mv0 = (const float*)d_in[2];
  const float* Wk  = (const float*)d_in[3];
  const float* bk  = (const float*)d_in[4];
  const float* Wv  = (const float*)d_in[5];
  const float* bv  = (const float*)d_in[6];
  const float* W1  = (const float*)d_in[7];
  const float* b1  = (const float*)d_in[8];
  const float* W2  = (const float*)d_in[9];
  const float* b2  = (const float*)d_in[10];
  float* out = (float*)d_out;

  // workspace layout (floats): mutable memory copies (128 MB, L2-resident) + small state
  float* ws = (float*)d_ws;
  const size_t MKF = (size_t)NSLOT * DDIM;
  float* mk     = ws;
  float* mv     = ws + MKF;
  float* Kall   = ws + 2 * MKF;
  float* WVall  = Kall + (size_t)BATCH * DDIM;
  float* scores = WVall + (size_t)BATCH * DDIM;
  float* rv     = scores + NSLOT;
  float* h      = rv + DDIM;
  float* wsum   = h + HDIM;
  unsigned long long* maxpack = (unsigned long long*)(wsum + 2);  // 8B aligned
  unsigned* cnt = (unsigned*)(maxpack + 1);
  unsigned* gen = cnt + 1;

  // inputs must not be mutated -> copy memory state into workspace each call
  hipMemcpyAsync(mk, mk0, MKF * sizeof(float), hipMemcpyDeviceToDevice, stream);
  hipMemcpyAsync(mv, mv0, MKF * sizeof(float), hipMemcpyDeviceToDevice, stream);
  const size_t zbytes = (size_t)((char*)(gen + 1) - (char*)rv);
  hipMemsetAsync(rv, 0, zbytes, stream);  // rv, wsum, maxpack, barrier cnt/gen

  proj_gemm<<<NBLK, TPB, 0, stream>>>(S, Wk, bk, Wv, bv, Kall, WVall);
  memsim_persistent<<<NBLK, TPB, 0, stream>>>(mk, mv, Kall, WVall, S, W1, b1, W2, b2,
                                              scores, rv, h, wsum, maxpack, cnt, gen,
                                              out);
}